// GAT_4449586118921
// MI455X (gfx1250) — compile-verified
//
#include <hip/hip_runtime.h>
#include <hip/hip_bf16.h>
#include <cstdint>
#include <cstddef>

#define NN    50000
#define EE    800000
#define ETOT  (EE + NN)     // edges + self loops = 850000
#define FIN   256
#define HC1   128           // HEADS*HID
#define H1    4
#define C1    32
#define OUTC  40
#define NEGSL 0.2f
#define BNEPS 1e-5f

typedef float v2f __attribute__((ext_vector_type(2)));
typedef float v8f __attribute__((ext_vector_type(8)));

// ---------- order-preserving float <-> uint encoding for atomicMax ----------
__device__ __forceinline__ unsigned fenc(float f) {
    unsigned u = __float_as_uint(f);
    return (u & 0x80000000u) ? ~u : (u | 0x80000000u);
}
__device__ __forceinline__ float fdec(unsigned u) {
    u = (u & 0x80000000u) ? (u & 0x7fffffffu) : ~u;
    return __uint_as_float(u);
}

// ---------- GEMM1: h1[N,128] = X[N,256] @ W1[256,128], f32 WMMA ----------
__global__ void gemm1_wmma(const float* __restrict__ X,
                           const float* __restrict__ W,
                           float* __restrict__ Hout) {
    const int lane = threadIdx.x & 31;
    const int wave = threadIdx.x >> 5;
    const int half = lane >> 4;       // K-pair selector (ISA 16x4 f32 A layout)
    const int r    = lane & 15;       // M for A, N for B/C/D
    const int m0   = blockIdx.x * 16;
    const int n0   = (blockIdx.y * 4 + wave) * 16;   // 8 N-tiles of 16 = 128

    const float* xrow = X + (size_t)(m0 + r) * FIN;
    v8f acc = {0.f, 0.f, 0.f, 0.f, 0.f, 0.f, 0.f, 0.f};
    for (int k = 0; k < FIN; k += 4) {
        const int ka = k + 2 * half;
        v2f a, b;
        a[0] = xrow[ka];
        a[1] = xrow[ka + 1];
        b[0] = W[(size_t)ka * HC1 + n0 + r];
        b[1] = W[(size_t)(ka + 1) * HC1 + n0 + r];
        acc = __builtin_amdgcn_wmma_f32_16x16x4_f32(false, a, false, b,
                                                    (short)0, acc, false, false);
    }
#pragma unroll
    for (int g = 0; g < 8; ++g)
        Hout[(size_t)(m0 + g + 8 * half) * HC1 + n0 + r] = acc[g];
}

// ---------- GEMM2: h2[N,40] = Hbn[N,128] @ W2[128,40], f32 WMMA ----------
__global__ void gemm2_wmma(const float* __restrict__ A,
                           const float* __restrict__ W,
                           float* __restrict__ Out) {
    const int lane = threadIdx.x & 31;
    const int wave = threadIdx.x >> 5;
    const int half = lane >> 4;
    const int r    = lane & 15;
    const int mt   = blockIdx.x * 4 + wave;
    if (mt >= NN / 16) return;                 // wave-uniform exit
    const int m0  = mt * 16;
    const int n0  = blockIdx.y * 16;           // 3 tiles -> 48 cols, guard at 40
    const int col = n0 + r;

    const float* arow = A + (size_t)(m0 + r) * HC1;
    v8f acc = {0.f, 0.f, 0.f, 0.f, 0.f, 0.f, 0.f, 0.f};
    for (int k = 0; k < HC1; k += 4) {
        const int ka = k + 2 * half;
        v2f a, b;
        a[0] = arow[ka];
        a[1] = arow[ka + 1];
        b[0] = (col < OUTC) ? W[(size_t)ka * OUTC + col] : 0.f;
        b[1] = (col < OUTC) ? W[(size_t)(ka + 1) * OUTC + col] : 0.f;
        acc = __builtin_amdgcn_wmma_f32_16x16x4_f32(false, a, false, b,
                                                    (short)0, acc, false, false);
    }
    if (col < OUTC) {
#pragma unroll
        for (int g = 0; g < 8; ++g)
            Out[(size_t)(m0 + g + 8 * half) * OUTC + col] = acc[g];
    }
}

// ---------- attention dots: a_src[n,h], a_dst[n,h] ----------
template <int H, int C>
__global__ void attdot(const float* __restrict__ hfeat,
                       const float* __restrict__ attS,
                       const float* __restrict__ attD,
                       float* __restrict__ aS, float* __restrict__ aD, int n) {
    int idx = blockIdx.x * blockDim.x + threadIdx.x;
    if (idx >= n * H) return;
    int node = idx / H, h = idx - node * H;
    const float* row = hfeat + (size_t)node * (H * C) + h * C;
    float s = 0.f, d = 0.f;
#pragma unroll 8
    for (int c = 0; c < C; ++c) {
        float v = row[c];
        s += v * attS[h * C + c];
        d += v * attD[h * C + c];
    }
    aS[idx] = s;
    aD[idx] = d;
}

// ---------- edge logits + segment max ----------
template <int H>
__global__ void edge_logits(const int* __restrict__ src, const int* __restrict__ dst,
                            const float* __restrict__ aS, const float* __restrict__ aD,
                            float* __restrict__ e, unsigned* __restrict__ m) {
    int idx = blockIdx.x * blockDim.x + threadIdx.x;
    if (idx >= ETOT * H) return;
    int ed = idx / H, h = idx - ed * H;
    int s = (ed < EE) ? src[ed] : (ed - EE);
    int d = (ed < EE) ? dst[ed] : (ed - EE);
    float v = aS[s * H + h] + aD[d * H + h];
    v = (v > 0.f) ? v : NEGSL * v;
    e[idx] = v;
    atomicMax(&m[d * H + h], fenc(v));
}

// ---------- exp(e - max) + segment sum ----------
template <int H>
__global__ void edge_exp(const int* __restrict__ dst,
                         float* __restrict__ e, const unsigned* __restrict__ m,
                         float* __restrict__ ssum) {
    int idx = blockIdx.x * blockDim.x + threadIdx.x;
    if (idx >= ETOT * H) return;
    int ed = idx / H, h = idx - ed * H;
    int d = (ed < EE) ? dst[ed] : (ed - EE);
    float ex = expf(e[idx] - fdec(m[d * H + h]));
    e[idx] = ex;
    atomicAdd(&ssum[d * H + h], ex);
}

// ---------- weighted scatter: out[dst] += h[src] * alpha ----------
template <int H, int C, int TPE>
__global__ void aggregate(const int* __restrict__ src, const int* __restrict__ dst,
                          const float* __restrict__ hfeat,
                          const float* __restrict__ ex, const float* __restrict__ ssum,
                          float* __restrict__ out) {
    long long gid = (long long)blockIdx.x * blockDim.x + threadIdx.x;
    int ed = (int)(gid / TPE);
    int j  = (int)(gid - (long long)ed * TPE);
    if (ed >= ETOT || j >= H * C) return;
    int s = (ed < EE) ? src[ed] : (ed - EE);
    int d = (ed < EE) ? dst[ed] : (ed - EE);
    int h = j / C;
    float alpha = ex[(size_t)ed * H + h] / (ssum[d * H + h] + 1e-16f);
    atomicAdd(&out[(size_t)d * (H * C) + j], hfeat[(size_t)s * (H * C) + j] * alpha);
}

// ---------- BatchNorm stats (sum / sumsq per feature) ----------
__global__ void bn_stats(const float* __restrict__ hcat, float* __restrict__ sums) {
    int f = threadIdx.x;                      // 128 features
    int rowsPerBlk = (NN + gridDim.x - 1) / gridDim.x;
    int r0 = blockIdx.x * rowsPerBlk;
    int r1 = min(r0 + rowsPerBlk, NN);
    float s = 0.f, s2 = 0.f;
    for (int r = r0; r < r1; ++r) {
        float v = hcat[(size_t)r * HC1 + f];  // coalesced: lane==feature
        s  += v;
        s2 += v * v;
    }
    atomicAdd(&sums[f], s);
    atomicAdd(&sums[HC1 + f], s2);
}

// ---------- BatchNorm apply + ELU (in place) ----------
__global__ void bn_apply_elu(float* __restrict__ hcat, const float* __restrict__ sums,
                             const float* __restrict__ gamma, const float* __restrict__ beta) {
    int idx = blockIdx.x * blockDim.x + threadIdx.x;
    if (idx >= NN * HC1) return;
    int f = idx & (HC1 - 1);
    const float invN = 1.0f / (float)NN;
    float mu  = sums[f] * invN;
    float var = sums[HC1 + f] * invN - mu * mu;
    float y = (hcat[idx] - mu) * rsqrtf(var + BNEPS) * gamma[f] + beta[f];
    hcat[idx] = (y > 0.f) ? y : expm1f(y);    // ELU(alpha=1)
}

// ---------- final bias add on output ----------
__global__ void add_bias(float* __restrict__ out, const float* __restrict__ b) {
    int idx = blockIdx.x * blockDim.x + threadIdx.x;
    if (idx >= NN * OUTC) return;
    out[idx] += b[idx % OUTC];
}

extern "C" void kernel_launch(void* const* d_in, const int* in_sizes, int n_in,
                              void* d_out, int out_size, void* d_ws, size_t ws_size,
                              hipStream_t stream) {
    (void)in_sizes; (void)n_in; (void)out_size; (void)ws_size;

    const float* x     = (const float*)d_in[0];
    const int*   ei    = (const int*)d_in[1];        // [2,E] int32
    const float* W1    = (const float*)d_in[2];
    const float* attS1 = (const float*)d_in[3];
    const float* attD1 = (const float*)d_in[4];
    // d_in[5] = b1 : cancelled exactly by training-mode BatchNorm
    const float* gamma = (const float*)d_in[6];
    const float* beta  = (const float*)d_in[7];
    const float* W2    = (const float*)d_in[8];
    const float* attS2 = (const float*)d_in[9];
    const float* attD2 = (const float*)d_in[10];
    const float* b2    = (const float*)d_in[11];
    float* out = (float*)d_out;
    char*  ws  = (char*)d_ws;

    // workspace layout (bytes)
    float*    h1    = (float*)(ws + 0);           // N*128*4 = 25,600,000
    float*    agg1  = (float*)(ws + 25600000);    // N*128*4
    float*    e1    = (float*)(ws + 51200000);    // ETOT*4*4 = 13,600,000
    float*    asrc1 = (float*)(ws + 64800000);    // N*4*4
    float*    adst1 = (float*)(ws + 65600000);
    unsigned* m1    = (unsigned*)(ws + 66400000);
    float*    s1    = (float*)(ws + 67200000);
    float*    sums  = (float*)(ws + 68000000);    // 256 floats
    // layer-2 buffers reuse h1 / e1 regions (dead after aggregate1 / BN)
    float*    h2    = (float*)(ws + 0);           // N*40*4 = 8,000,000
    float*    e2    = (float*)(ws + 51200000);    // ETOT*4 = 3,400,000
    float*    asrc2 = (float*)(ws + 54600000);
    float*    adst2 = (float*)(ws + 54800000);
    unsigned* m2    = (unsigned*)(ws + 55000000);
    float*    s2    = (float*)(ws + 55200000);

    const int* srcI = ei;
    const int* dstI = ei + EE;

    // ---- layer 1 ----
    hipMemsetAsync(agg1, 0, (size_t)NN * HC1 * 4, stream);
    hipMemsetAsync(m1,   0, (size_t)NN * H1 * 4, stream);
    hipMemsetAsync(s1,   0, (size_t)NN * H1 * 4, stream);
    hipMemsetAsync(sums, 0, 2 * HC1 * 4, stream);

    gemm1_wmma<<<dim3(NN / 16, 2), 128, 0, stream>>>(x, W1, h1);
    attdot<H1, C1><<<(NN * H1 + 255) / 256, 256, 0, stream>>>(h1, attS1, attD1, asrc1, adst1, NN);
    edge_logits<H1><<<(ETOT * H1 + 255) / 256, 256, 0, stream>>>(srcI, dstI, asrc1, adst1, e1, m1);
    edge_exp<H1><<<(ETOT * H1 + 255) / 256, 256, 0, stream>>>(dstI, e1, m1, s1);
    aggregate<H1, C1, 128><<<(unsigned)(((long long)ETOT * 128 + 255) / 256), 256, 0, stream>>>(
        srcI, dstI, h1, e1, s1, agg1);
    bn_stats<<<512, HC1, 0, stream>>>(agg1, sums);
    bn_apply_elu<<<(NN * HC1 + 255) / 256, 256, 0, stream>>>(agg1, sums, gamma, beta);

    // ---- layer 2 ----
    hipMemsetAsync(m2, 0, (size_t)NN * 4, stream);
    hipMemsetAsync(s2, 0, (size_t)NN * 4, stream);
    hipMemsetAsync(out, 0, (size_t)NN * OUTC * 4, stream);

    gemm2_wmma<<<dim3((NN / 16 + 3) / 4, 3), 128, 0, stream>>>(agg1, W2, h2);
    attdot<1, OUTC><<<(NN + 255) / 256, 256, 0, stream>>>(h2, attS2, attD2, asrc2, adst2, NN);
    edge_logits<1><<<(ETOT + 255) / 256, 256, 0, stream>>>(srcI, dstI, asrc2, adst2, e2, m2);
    edge_exp<1><<<(ETOT + 255) / 256, 256, 0, stream>>>(dstI, e2, m2, s2);
    aggregate<1, OUTC, 64><<<(unsigned)(((long long)ETOT * 64 + 255) / 256), 256, 0, stream>>>(
        srcI, dstI, h2, e2, s2, out);
    add_bias<<<(NN * OUTC + 255) / 256, 256, 0, stream>>>(out, b2);
}